// GAT_37108517437513
// MI455X (gfx1250) — compile-verified
//
#include <hip/hip_runtime.h>

typedef __attribute__((ext_vector_type(2))) float v2f;
typedef __attribute__((ext_vector_type(8))) float v8f;

#define NEG_SLOPE 0.2f
#define GAT_EPS 1e-16f

// ---------- helpers ----------

__device__ __forceinline__ float leaky02(float v) {
  return v > 0.0f ? v : NEG_SLOPE * v;
}

// float atomic max via int tricks (valid with -inf initialization, mixed signs OK)
__device__ __forceinline__ void atomic_max_f32(float* addr, float val) {
  if (val >= 0.0f)
    atomicMax((int*)addr, __float_as_int(val));
  else
    atomicMin((unsigned int*)addr, (unsigned int)__float_as_int(val));
}

// src/dst for edge e; edges E..E+N-1 are the implicit self-loops
__device__ __forceinline__ void edge_sd(const int* __restrict__ ei, long long e,
                                        int E, int& s, int& d) {
  if (e < (long long)E) { s = ei[e]; d = ei[(long long)E + e]; }
  else { s = d = (int)(e - E); }
}

// ---------- GEMM: C[n_rows x n_cols] = A[n_rows x K] * B[K x n_cols] ----------
// grid.x = ceil(n_rows/16); blockDim.x = 32 * ceil(n_cols/16); one wave per 16-col tile.
// Uses V_WMMA_F32_16X16X4_F32 (fp32 in, fp32 accum) — K stepped by 4.

__global__ void wmma_gemm_f32(const float* __restrict__ A, const float* __restrict__ B,
                              float* __restrict__ C, int n_rows, int K, int n_cols) {
  const int lane = threadIdx.x & 31;
  const int wave = threadIdx.x >> 5;
  const int row_base = blockIdx.x << 4;
  const int col_base = wave << 4;
  if (col_base >= n_cols) return;            // uniform per wave: EXEC stays all-1s below

  const int m    = lane & 15;
  const int half = lane >> 4;                // selects K pair {0,1} vs {2,3}

  const int  arow   = row_base + m;
  const int  bcol   = col_base + m;
  const bool row_ok = arow < n_rows;
  const bool col_ok = bcol < n_cols;
  const int  arow_c = row_ok ? arow : 0;     // clamp so loads are always in-bounds
  const int  bcol_c = col_ok ? bcol : 0;

  const float* Ap = A + (long long)arow_c * K;

  v8f acc = {};
  for (int k = 0; k < K; k += 4) {
    const int ka = k + (half << 1);
    v2f a, b;
    float a0 = Ap[ka];
    float a1 = Ap[ka + 1];
    float b0 = B[(long long)ka * n_cols + bcol_c];
    float b1 = B[(long long)(ka + 1) * n_cols + bcol_c];
    a.x = row_ok ? a0 : 0.0f;  a.y = row_ok ? a1 : 0.0f;
    b.x = col_ok ? b0 : 0.0f;  b.y = col_ok ? b1 : 0.0f;
    acc = __builtin_amdgcn_wmma_f32_16x16x4_f32(false, a, false, b,
                                                (short)0, acc, false, false);
  }
#pragma unroll
  for (int r = 0; r < 8; ++r) {
    const int orow = row_base + r + (half << 3);
    if (orow < n_rows && col_ok)
      C[(long long)orow * n_cols + bcol] = acc[r];
  }
}

// ---------- per-(node,head) attention logits ----------

__global__ void alpha_kernel(const float* __restrict__ h,
                             const float* __restrict__ att_s,
                             const float* __restrict__ att_d,
                             float* __restrict__ asrc, float* __restrict__ adst,
                             int N, int H, int D) {
  long long idx = (long long)blockIdx.x * blockDim.x + threadIdx.x;
  if (idx >= (long long)N * H) return;
  int hh = (int)(idx % H);
  long long n = idx / H;
  const float* hp = h + (n * H + hh) * D;
  const float* as = att_s + (long long)hh * D;
  const float* ad = att_d + (long long)hh * D;
  float s = 0.0f, d = 0.0f;
  for (int i = 0; i < D; ++i) { float v = hp[i]; s += v * as[i]; d += v * ad[i]; }
  asrc[idx] = s; adst[idx] = d;
}

// ---------- init kernels ----------

__global__ void init_softmax(float* __restrict__ m, float* __restrict__ s, long long n) {
  long long i = (long long)blockIdx.x * blockDim.x + threadIdx.x;
  if (i >= n) return;
  m[i] = -__builtin_huge_valf();
  s[i] = 0.0f;
}

__global__ void init_out_bias(float* __restrict__ out, const float* __restrict__ bias,
                              long long n_elem, int cols) {
  long long i = (long long)blockIdx.x * blockDim.x + threadIdx.x;
  if (i >= n_elem) return;
  out[i] = bias[(int)(i % cols)];
}

// ---------- edge passes ----------

__global__ void edge_pass_max(const int* __restrict__ ei, int E, int N, int H,
                              const float* __restrict__ asrc, const float* __restrict__ adst,
                              float* __restrict__ mbuf) {
  long long idx = (long long)blockIdx.x * blockDim.x + threadIdx.x;
  long long total = (long long)(E + N) * H;
  if (idx >= total) return;
  int h = (int)(idx % H); long long e = idx / H;
  int s, d; edge_sd(ei, e, E, s, d);
  float v = leaky02(asrc[(long long)s * H + h] + adst[(long long)d * H + h]);
  atomic_max_f32(&mbuf[(long long)d * H + h], v);
}

__global__ void edge_pass_expsum(const int* __restrict__ ei, int E, int N, int H,
                                 const float* __restrict__ asrc, const float* __restrict__ adst,
                                 const float* __restrict__ mbuf, float* __restrict__ sbuf) {
  long long idx = (long long)blockIdx.x * blockDim.x + threadIdx.x;
  long long total = (long long)(E + N) * H;
  if (idx >= total) return;
  int h = (int)(idx % H); long long e = idx / H;
  int s, d; edge_sd(ei, e, E, s, d);
  long long dh = (long long)d * H + h;
  float v = leaky02(asrc[(long long)s * H + h] + adst[dh]);
  atomicAdd(&sbuf[dh], __expf(v - mbuf[dh]));
}

__global__ void edge_pass_agg(const int* __restrict__ ei, int E, int N, int H, int D,
                              const float* __restrict__ asrc, const float* __restrict__ adst,
                              const float* __restrict__ mbuf, const float* __restrict__ sbuf,
                              const float* __restrict__ hsrc, float* __restrict__ out) {
  long long idx = (long long)blockIdx.x * blockDim.x + threadIdx.x;
  long long total = (long long)(E + N) * H;
  if (idx >= total) return;
  int h = (int)(idx % H); long long e = idx / H;
  int s, d; edge_sd(ei, e, E, s, d);
  long long dh = (long long)d * H + h;
  float v = leaky02(asrc[(long long)s * H + h] + adst[dh]);
  float alpha = __expf(v - mbuf[dh]) / (sbuf[dh] + GAT_EPS);
  const float* hp = hsrc + ((long long)s * H + h) * D;
  float* op = out + dh * D;
  for (int i = 0; i < D; ++i)
    atomicAdd(&op[i], hp[i] * alpha);
}

// ---------- ELU (in place; input already has bias added via init_out_bias) ----------

__global__ void elu_kernel(float* __restrict__ x, long long n) {
  long long i = (long long)blockIdx.x * blockDim.x + threadIdx.x;
  if (i >= n) return;
  float v = x[i];
  x[i] = v > 0.0f ? v : (__expf(v) - 1.0f);
}

// ---------- host launch ----------

extern "C" void kernel_launch(void* const* d_in, const int* in_sizes, int n_in,
                              void* d_out, int out_size, void* d_ws, size_t ws_size,
                              hipStream_t stream) {
  const float* x      = (const float*)d_in[0];
  const int*   ei     = (const int*)  d_in[1];
  // d_in[2] = edge_attr (unused by reference math)
  const float* W1     = (const float*)d_in[3];
  const float* att_s1 = (const float*)d_in[4];
  const float* att_d1 = (const float*)d_in[5];
  const float* bias1  = (const float*)d_in[6];
  const float* W2     = (const float*)d_in[7];
  const float* att_s2 = (const float*)d_in[8];
  const float* att_d2 = (const float*)d_in[9];
  const float* bias2  = (const float*)d_in[10];
  float* out = (float*)d_out;

  const int F_IN = 128;
  const int N  = in_sizes[0] / F_IN;      // 50000
  const int E  = in_sizes[1] / 2;         // 1600000
  const int H1 = 8;
  const int HD = in_sizes[4];             // heads*out_dim = 64
  const int D1 = HD / H1;                 // 8
  const int C  = in_sizes[7] / HD;        // 40
  const int Et = E + N;                   // with self loops

  // workspace carve-up (floats)
  float* ws    = (float*)d_ws;
  float* h1    = ws;  ws += (size_t)N * HD;
  float* asrc1 = ws;  ws += (size_t)N * H1;
  float* adst1 = ws;  ws += (size_t)N * H1;
  float* m1    = ws;  ws += (size_t)N * H1;
  float* s1    = ws;  ws += (size_t)N * H1;
  float* out1  = ws;  ws += (size_t)N * HD;   // agg+bias, then ELU in place
  float* h2    = ws;  ws += (size_t)N * C;
  float* asrc2 = ws;  ws += (size_t)N;
  float* adst2 = ws;  ws += (size_t)N;
  float* m2    = ws;  ws += (size_t)N;
  float* s2    = ws;  ws += (size_t)N;

  const int row_tiles = (N + 15) / 16;
  const int B256 = 256;

  // ===== Layer 1 =====
  wmma_gemm_f32<<<dim3(row_tiles), dim3(32 * ((HD + 15) / 16)), 0, stream>>>(
      x, W1, h1, N, F_IN, HD);

  long long nh1 = (long long)N * H1;
  alpha_kernel<<<(unsigned)((nh1 + B256 - 1) / B256), B256, 0, stream>>>(
      h1, att_s1, att_d1, asrc1, adst1, N, H1, D1);
  init_softmax<<<(unsigned)((nh1 + B256 - 1) / B256), B256, 0, stream>>>(m1, s1, nh1);

  long long tE1 = (long long)Et * H1;
  unsigned gE1 = (unsigned)((tE1 + B256 - 1) / B256);
  edge_pass_max   <<<gE1, B256, 0, stream>>>(ei, E, N, H1, asrc1, adst1, m1);
  edge_pass_expsum<<<gE1, B256, 0, stream>>>(ei, E, N, H1, asrc1, adst1, m1, s1);

  long long nhd = (long long)N * HD;
  init_out_bias<<<(unsigned)((nhd + B256 - 1) / B256), B256, 0, stream>>>(out1, bias1, nhd, HD);
  edge_pass_agg<<<gE1, B256, 0, stream>>>(ei, E, N, H1, D1, asrc1, adst1, m1, s1, h1, out1);
  elu_kernel<<<(unsigned)((nhd + B256 - 1) / B256), B256, 0, stream>>>(out1, nhd);

  // ===== Layer 2 (heads = 1) =====
  wmma_gemm_f32<<<dim3(row_tiles), dim3(32 * ((C + 15) / 16)), 0, stream>>>(
      out1, W2, h2, N, HD, C);

  alpha_kernel<<<(unsigned)((N + B256 - 1) / B256), B256, 0, stream>>>(
      h2, att_s2, att_d2, asrc2, adst2, N, 1, C);
  init_softmax<<<(unsigned)((N + B256 - 1) / B256), B256, 0, stream>>>(m2, s2, N);

  long long tE2 = (long long)Et;
  unsigned gE2 = (unsigned)((tE2 + B256 - 1) / B256);
  edge_pass_max   <<<gE2, B256, 0, stream>>>(ei, E, N, 1, asrc2, adst2, m2);
  edge_pass_expsum<<<gE2, B256, 0, stream>>>(ei, E, N, 1, asrc2, adst2, m2, s2);

  long long nc = (long long)N * C;
  init_out_bias<<<(unsigned)((nc + B256 - 1) / B256), B256, 0, stream>>>(out, bias2, nc, C);
  edge_pass_agg<<<gE2, B256, 0, stream>>>(ei, E, N, 1, C, asrc2, adst2, m2, s2, h2, out);
}